// QuantumGenerator_78537771974891
// MI455X (gfx1250) — compile-verified
//
#include <hip/hip_runtime.h>
#include <hip/hip_bf16.h>

// ---------------------------------------------------------------------------
// 14-qubit batched state-vector simulator, fully LDS-resident per sample.
//
// Strategy (MI455X / gfx1250):
//  * 1 workgroup per batch sample (64 WGs), 512 threads = 16 wave32 waves.
//  * State (16384 complex = 128 KB) lives entirely in LDS (320 KB/WGP).
//  * All 98 gate matrices (14 encoding RX*RY + 6*14 variational RY) are
//    precomputed ONCE per block (one sincosf per thread) into an LDS table;
//    the hot passes only do uniform ds_load broadcasts — no transcendentals.
//  * Single-qubit gates are fused 4-at-a-time into a 16x16 unitary
//    (Kronecker product) and applied as D = U * S_tile with chained
//    V_WMMA_F32_16X16X4_F32 (complex matmul = 2..4 real matmuls).
//  * Tile loop has a static trip count (4) so it lowers without EXEC masking
//    and can be unrolled to overlap ds_load of tile i+1 with WMMAs of tile i.
//  * CZ ladder of a layer fuses into one sign pass: parity of popc(i & i>>1).
//  * Expectations <X_w> reduced wave-wide (shfl, wave32) then across waves.
// ---------------------------------------------------------------------------

#define NQ      14
#define DIM     (1 << NQ)     // 16384 amplitudes
#define BATCH   64
#define QDEPTH  6
#define THREADS 512
#define NWAVES  (THREADS / 32)
#define NTILES  (64 / NWAVES)         // 4 tiles per wave per pass
#define NGATES  ((1 + QDEPTH) * NQ)   // 98 gates: pass 0 = encoding, 1..6 = RY

typedef __attribute__((ext_vector_type(2))) float v2f;
typedef __attribute__((ext_vector_type(8))) float v8f;

__device__ __forceinline__ float2 cmul(float2 a, float2 b) {
    return make_float2(a.x * b.x - a.y * b.y, a.x * b.y + a.y * b.x);
}

// Select g[mb][kb] with runtime bits using only constant private indexing
// (keeps the 2x2 gates in VGPRs, selection lowers to v_cndmask).
__device__ __forceinline__ float2 pick(const float2 g[2][2], int mb, int kb) {
    float2 r0 = kb ? g[0][1] : g[0][0];
    float2 r1 = kb ? g[1][1] : g[1][0];
    return mb ? r1 : r0;
}

__device__ __forceinline__ void set_ident(float2 g[2][2]) {
    g[0][0] = make_float2(1.f, 0.f); g[0][1] = make_float2(0.f, 0.f);
    g[1][0] = make_float2(0.f, 0.f); g[1][1] = make_float2(1.f, 0.f);
}

// Encoding gate: M = RX(t) * RY(t)  (RY applied first in the reference).
__device__ __forceinline__ void set_enc(float t, float2 g[2][2]) {
    float s, c;
    sincosf(0.5f * t, &s, &c);
    g[0][0] = make_float2(c * c, -s * s);
    g[0][1] = make_float2(-s * c, -s * c);
    g[1][0] = make_float2(s * c, -s * c);
    g[1][1] = make_float2(c * c, s * s);
}

// Variational gate: RY(t), purely real.
__device__ __forceinline__ void set_ry(float t, float2 g[2][2]) {
    float s, c;
    sincosf(0.5f * t, &s, &c);
    g[0][0] = make_float2(c, 0.f); g[0][1] = make_float2(-s, 0.f);
    g[1][0] = make_float2(s, 0.f); g[1][1] = make_float2(c, 0.f);
}

// Uniform-address broadcast load of a precomputed 2x2 gate from LDS.
__device__ __forceinline__ void load_gate(const float2* gbuf, int gid,
                                          float2 g[2][2]) {
    const float2* p = gbuf + gid * 4;
    g[0][0] = p[0]; g[0][1] = p[1];
    g[1][0] = p[2]; g[1][1] = p[3];
}

// Apply U = g0 (x) g1 (x) g2 (x) g3 (16x16) to the 4-bit field at bit offset
// `shift` of the amplitude index:  S'[m, c] = sum_k U[m][k] * S[k, c].
// g0 acts on bit shift+3 (field bit 3), g3 on bit shift (field bit 0).
// Each wave computes 4 disjoint 16-column tiles via chained WMMA f32 16x16x4.
//
// Fragment layouts (CDNA5 ISA 7.12.2, wave32):
//   A (16x4 f32, 2 VGPRs): lane L -> M = L&15; VGPR v -> K = k0 + 2*(L>>4) + v
//   B ( 4x16 f32, 2 VGPRs): lane L -> N = L&15; VGPR v -> K = k0 + 2*(L>>4) + v
//   C/D (16x16 f32, 8 VGPRs): lane L -> N = L&15; VGPR v -> M = v + 8*(L>>4)
__device__ __forceinline__ void apply_group(float* __restrict__ sRe,
                                            float* __restrict__ sIm,
                                            const float2 (&g)[4][2][2],
                                            int shift, bool complexU) {
    const int lane = threadIdx.x & 31;
    const int wid  = threadIdx.x >> 5;
    const int n    = lane & 15;   // column within tile / A-row index
    const int h    = lane >> 4;   // lane-half
    const int m    = n;           // A-matrix row carried by this lane

    // Build A fragments: this lane holds U[m][k] for k = 4q + 2h + v.
    v2f aRe[4], aIm[4], aImN[4];
#pragma unroll
    for (int q = 0; q < 4; ++q) {
#pragma unroll
        for (int v = 0; v < 2; ++v) {
            const int k = 4 * q + 2 * h + v;
            float2 p = pick(g[0], (m >> 3) & 1, (k >> 3) & 1);
            p = cmul(p, pick(g[1], (m >> 2) & 1, (k >> 2) & 1));
            p = cmul(p, pick(g[2], (m >> 1) & 1, (k >> 1) & 1));
            p = cmul(p, pick(g[3], m & 1, k & 1));
            if (v == 0) { aRe[q].x = p.x; aIm[q].x = p.y; aImN[q].x = -p.y; }
            else        { aRe[q].y = p.x; aIm[q].y = p.y; aImN[q].y = -p.y; }
        }
    }

    const int lowmask = (1 << shift) - 1;

    // 64 tiles of 16 columns; tiles are disjoint across waves -> no hazard.
    // Static trip count: no EXEC-masked loop, and unrolling lets the next
    // tile's ds_loads issue under the current tile's WMMA chain.
#pragma unroll 2
    for (int it = 0; it < NTILES; ++it) {
        const int t    = wid + it * NWAVES;
        const int cc   = t * 16 + n;                                   // column
        const int base = ((cc >> shift) << (shift + 4)) | (cc & lowmask);

        // Gather B fragments from LDS: B[k][n] = S[base + (k << shift)].
        v2f bRe[4], bIm[4];
#pragma unroll
        for (int q = 0; q < 4; ++q) {
            const int k0 = 4 * q + 2 * h;
            const int i0 = base + (k0 << shift);
            const int i1 = base + ((k0 + 1) << shift);
            bRe[q].x = sRe[i0]; bRe[q].y = sRe[i1];
            bIm[q].x = sIm[i0]; bIm[q].y = sIm[i1];
        }

        v8f dre = {0.f, 0.f, 0.f, 0.f, 0.f, 0.f, 0.f, 0.f};
        v8f dim = {0.f, 0.f, 0.f, 0.f, 0.f, 0.f, 0.f, 0.f};

        // Re(D) = Ur*Br - Ui*Bi ; Im(D) = Ur*Bi + Ui*Br  (chained over K).
#pragma unroll
        for (int q = 0; q < 4; ++q)
            dre = __builtin_amdgcn_wmma_f32_16x16x4_f32(
                false, aRe[q], false, bRe[q], (short)0, dre, false, false);
#pragma unroll
        for (int q = 0; q < 4; ++q)
            dim = __builtin_amdgcn_wmma_f32_16x16x4_f32(
                false, aRe[q], false, bIm[q], (short)0, dim, false, false);
        if (complexU) {
#pragma unroll
            for (int q = 0; q < 4; ++q)
                dre = __builtin_amdgcn_wmma_f32_16x16x4_f32(
                    false, aImN[q], false, bIm[q], (short)0, dre, false, false);
#pragma unroll
            for (int q = 0; q < 4; ++q)
                dim = __builtin_amdgcn_wmma_f32_16x16x4_f32(
                    false, aIm[q], false, bRe[q], (short)0, dim, false, false);
        }

        // Scatter D back: VGPR v holds row M = v + 8h, col n.
#pragma unroll
        for (int v = 0; v < 8; ++v) {
            const int M = v + 8 * h;
            const int a = base + (M << shift);
            sRe[a] = dre[v];
            sIm[a] = dim[v];
        }
    }
}

__global__ __launch_bounds__(THREADS) void qcircuit_kernel(
    const float* __restrict__ noise,   // [BATCH, NQ]
    const float* __restrict__ qp,      // [QDEPTH, NQ]
    float* __restrict__ out) {         // [BATCH, NQ]
    extern __shared__ float smem[];
    float*  sRe  = smem;                          // [DIM]
    float*  sIm  = smem + DIM;                    // [DIM]
    float*  red  = smem + 2 * DIM;                // [NWAVES] reduction scratch
    float2* gbuf = (float2*)(smem + 2 * DIM + 32); // [NGATES*4] gate table

    const int b   = blockIdx.x;
    const int tid = threadIdx.x;

    // |0...0>  +  precompute the whole gate table (one sincosf per thread).
    for (int i = tid; i < DIM; i += THREADS) { sRe[i] = 0.f; sIm[i] = 0.f; }
    if (tid == 0) sRe[0] = 1.f;
    if (tid < NGATES) {
        const int pass = tid / NQ;        // 0 = encoding, 1..QDEPTH = layers
        const int w    = tid % NQ;
        float2 gt[2][2];
        if (pass == 0) set_enc(noise[b * NQ + w], gt);
        else           set_ry(qp[(pass - 1) * NQ + w], gt);
        float2* dst = gbuf + tid * 4;
        dst[0] = gt[0][0]; dst[1] = gt[0][1];
        dst[2] = gt[1][0]; dst[3] = gt[1][1];
    }
    __syncthreads();

    float2 g[4][2][2];

    // ---- encoding layer: fused RX*RY per wire, complex 16x16 unitaries ----
    // wires ws..ws+3 occupy bits (13-ws)..(10-ws)  =>  shift = 10 - ws
    for (int gi = 0; gi < 3; ++gi) {
        const int ws = gi * 4;
#pragma unroll
        for (int k = 0; k < 4; ++k) load_gate(gbuf, ws + k, g[k]);
        apply_group(sRe, sIm, g, 10 - ws, true);
        __syncthreads();
    }
    // wires 12,13 on bits 1,0; identity-pad bits 3,2.
    set_ident(g[0]); set_ident(g[1]);
    load_gate(gbuf, 12, g[2]);
    load_gate(gbuf, 13, g[3]);
    apply_group(sRe, sIm, g, 0, true);
    __syncthreads();

    // ---- variational layers: RY per wire (real 16x16), then CZ ladder ----
    for (int l = 0; l < QDEPTH; ++l) {
        const int gofs = (l + 1) * NQ;
        for (int gi = 0; gi < 3; ++gi) {
            const int ws = gi * 4;
#pragma unroll
            for (int k = 0; k < 4; ++k) load_gate(gbuf, gofs + ws + k, g[k]);
            apply_group(sRe, sIm, g, 10 - ws, false);
            __syncthreads();
        }
        set_ident(g[0]); set_ident(g[1]);
        load_gate(gbuf, gofs + 12, g[2]);
        load_gate(gbuf, gofs + 13, g[3]);
        apply_group(sRe, sIm, g, 0, false);
        __syncthreads();

        // All 13 CZs at once: sign flip iff #adjacent-11 bit pairs is odd.
        for (int i = tid; i < DIM; i += THREADS) {
            if (__popc(i & (i >> 1)) & 1) { sRe[i] = -sRe[i]; sIm[i] = -sIm[i]; }
        }
        __syncthreads();
    }

    // ---- <X_w> = 2 * sum Re(conj(psi_0) psi_1) over the wire split ----
    const int lane = tid & 31, wid = tid >> 5;
    for (int w = 0; w < NQ; ++w) {
        const int tb = NQ - 1 - w;
        const int R  = 1 << tb;
        float acc = 0.f;
        for (int p = tid; p < DIM / 2; p += THREADS) {
            const int i0 = ((p >> tb) << (tb + 1)) | (p & (R - 1));
            const int i1 = i0 + R;
            acc += sRe[i0] * sRe[i1] + sIm[i0] * sIm[i1];
        }
        // wave32 reduction
        for (int off = 16; off > 0; off >>= 1) acc += __shfl_down(acc, off, 32);
        if (lane == 0) red[wid] = acc;
        __syncthreads();
        if (wid == 0) {
            float v = (lane < NWAVES) ? red[lane] : 0.f;
            for (int off = NWAVES / 2; off > 0; off >>= 1)
                v += __shfl_down(v, off, 32);
            if (lane == 0) out[b * NQ + w] = 2.f * v;
        }
        __syncthreads();
    }
}

extern "C" void kernel_launch(void* const* d_in, const int* in_sizes, int n_in,
                              void* d_out, int out_size, void* d_ws,
                              size_t ws_size, hipStream_t stream) {
    const float* noise = (const float*)d_in[0];  // [64, 14]
    const float* qp    = (const float*)d_in[1];  // [6, 14]
    float* out         = (float*)d_out;          // [64, 14]
    (void)in_sizes; (void)n_in; (void)out_size; (void)d_ws; (void)ws_size;

    // state (128 KB) + reduction pad + gate table (98 * 4 * float2)
    const size_t shmem = (2 * DIM + 32) * sizeof(float)
                       + NGATES * 4 * sizeof(float2);
    qcircuit_kernel<<<BATCH, THREADS, shmem, stream>>>(noise, qp, out);
}